// MultiMLP_65601330479166
// MI455X (gfx1250) — compile-verified
//
#include <hip/hip_runtime.h>
#include <hip/hip_bf16.h>

typedef __attribute__((ext_vector_type(2))) float v2f;
typedef __attribute__((ext_vector_type(8))) float v8f;

#define WIN    5
#define TLEN   1000
#define WC     996
#define HDIM   256
#define CDIM   256
#define KB     16      // K-panel depth
#define PROW   20      // panel row stride (dwords), pad=4 -> bank-conflict free
#define HROW   260     // h row stride (dwords), pad=4 -> bank-conflict free

__global__ __launch_bounds__(256)
void multimlp_wmma_f32(const float* __restrict__ x,
                       const float* __restrict__ W1,
                       const float* __restrict__ b1,
                       const float* __restrict__ W2,
                       const float* __restrict__ b2,
                       float* __restrict__ out)
{
    const int w    = blockIdx.x;          // window index, 0..995
    const int tid  = threadIdx.x;         // 0..255
    const int lane = tid & 31;
    const int wave = tid >> 5;            // 0..7
    const int hi   = lane >> 4;           // 0/1 (upper half-lanes)
    const int lo   = lane & 15;

    __shared__ float sH[32 * HROW];       // layer-1 activations, 32 batch rows
    __shared__ float sW[256 * PROW];      // W2 K-panel: [c][k]
    __shared__ float sX[32 * 8];          // x windows for current batch half

    // per-thread layer-1 weights: thread tid owns hidden unit tid
    const float* w1p = W1 + ((size_t)w * HDIM + tid) * WIN;
    const float w10 = w1p[0], w11 = w1p[1], w12 = w1p[2], w13 = w1p[3], w14 = w1p[4];
    const float bias1 = b1[(size_t)w * HDIM + tid];

    const float* W2w = W2 + (size_t)w * CDIM * HDIM;   // [c][h], rows contiguous in h

    const int n0 = wave * 32;             // this wave's output-column block

    for (int half = 0; half < 2; ++half) {
        const int bbase = half * 32;

        // ---- stage x windows ----
        if (tid < 32) {
            const float* xp = x + (size_t)(bbase + tid) * TLEN + w;
            #pragma unroll
            for (int s = 0; s < WIN; ++s) sX[tid * 8 + s] = xp[s];
        }
        __syncthreads();

        // ---- layer 1: sH[b][tid] = relu(x·W1 + b1) ----
        #pragma unroll 8
        for (int b = 0; b < 32; ++b) {
            float acc = bias1;
            acc = fmaf(sX[b*8+0], w10, acc);
            acc = fmaf(sX[b*8+1], w11, acc);
            acc = fmaf(sX[b*8+2], w12, acc);
            acc = fmaf(sX[b*8+3], w13, acc);
            acc = fmaf(sX[b*8+4], w14, acc);
            sH[b * HROW + tid] = fmaxf(acc, 0.0f);
        }
        __syncthreads();

        // ---- layer 2: 32x32 tile per wave via f32 WMMA ----
        v8f acc00 = {}, acc01 = {}, acc10 = {}, acc11 = {};

        for (int p = 0; p < HDIM / KB; ++p) {
            const int k0 = p * KB;
            if (p != 0) __syncthreads();   // previous panel fully consumed

            // stage W2 panel: thread tid copies row c=tid, 16 floats (4 x b128)
            {
                const float4* src = (const float4*)(W2w + (size_t)tid * HDIM + k0);
                float4* dst = (float4*)(sW + tid * PROW);
                #pragma unroll
                for (int q = 0; q < 4; ++q) dst[q] = src[q];
            }
            if (p + 1 < HDIM / KB)
                __builtin_prefetch(W2w + (size_t)tid * HDIM + k0 + KB, 0, 3);
            __syncthreads();

            #pragma unroll
            for (int kq = 0; kq < KB / 4; ++kq) {
                const int kk = kq * 4 + 2 * hi;   // K offset held by this half-lane
                // A fragments: A[M=row][K], lane(hi,lo) holds rows lo, K = kk,kk+1
                v2f a0 = *(const v2f*)(sH + (0 * 16 + lo) * HROW + k0 + kk);
                v2f a1 = *(const v2f*)(sH + (1 * 16 + lo) * HROW + k0 + kk);
                // B fragments: B[K][N=c] = W2[c][k]; lane holds col n0+ntile*16+lo
                v2f bb0 = *(const v2f*)(sW + (n0 + 0 * 16 + lo) * PROW + kk);
                v2f bb1 = *(const v2f*)(sW + (n0 + 1 * 16 + lo) * PROW + kk);

                acc00 = __builtin_amdgcn_wmma_f32_16x16x4_f32(false, a0, false, bb0,
                                                              (short)0, acc00, false, false);
                acc01 = __builtin_amdgcn_wmma_f32_16x16x4_f32(false, a0, false, bb1,
                                                              (short)0, acc01, false, false);
                acc10 = __builtin_amdgcn_wmma_f32_16x16x4_f32(false, a1, false, bb0,
                                                              (short)0, acc10, false, false);
                acc11 = __builtin_amdgcn_wmma_f32_16x16x4_f32(false, a1, false, bb1,
                                                              (short)0, acc11, false, false);
            }
        }
        __syncthreads();  // guard sH/sW before next half rewrites them

        // ---- epilogue: add b2, store. D layout: VGPR j -> (M = j + 8*hi, N = lo) ----
        const int c0 = n0 + lo;
        const float bias2_0 = b2[(size_t)w * CDIM + c0];
        const float bias2_1 = b2[(size_t)w * CDIM + c0 + 16];
        #pragma unroll
        for (int j = 0; j < 8; ++j) {
            const int m0r = bbase + 0 * 16 + j + 8 * hi;
            const int m1r = bbase + 1 * 16 + j + 8 * hi;
            float* o0 = out + ((size_t)m0r * WC + w) * CDIM;
            float* o1 = out + ((size_t)m1r * WC + w) * CDIM;
            o0[c0]      = acc00[j] + bias2_0;
            o0[c0 + 16] = acc01[j] + bias2_1;
            o1[c0]      = acc10[j] + bias2_0;
            o1[c0 + 16] = acc11[j] + bias2_1;
        }
    }
}

extern "C" void kernel_launch(void* const* d_in, const int* in_sizes, int n_in,
                              void* d_out, int out_size, void* d_ws, size_t ws_size,
                              hipStream_t stream)
{
    (void)in_sizes; (void)n_in; (void)out_size; (void)d_ws; (void)ws_size;
    const float* x  = (const float*)d_in[0];
    const float* W1 = (const float*)d_in[1];
    const float* b1 = (const float*)d_in[2];
    const float* W2 = (const float*)d_in[3];
    const float* b2 = (const float*)d_in[4];
    float* out = (float*)d_out;

    dim3 grid(WC), block(256);
    hipLaunchKernelGGL(multimlp_wmma_f32, grid, block, 0, stream,
                       x, W1, b1, W2, b2, out);
}